// PositionsToGraphTransform_11768210391286
// MI455X (gfx1250) — compile-verified
//
#include <hip/hip_runtime.h>
#include <hip/hip_fp16.h>

// Problem constants (match reference)
#define BATCH   16
#define NPTS    4096
#define KNN     50
#define NBINS   5
#define TILE    16              // WMMA M/N tile
#define NTILES  (NPTS / TILE)   // 256 column tiles
#define WAVES   8               // 256-thread block = 8 wave32

typedef __attribute__((ext_vector_type(2)))  float        v2f;
typedef __attribute__((ext_vector_type(8)))  float        v8f;
typedef __attribute__((ext_vector_type(4)))  unsigned int v4u;
typedef __attribute__((ext_vector_type(4)))  int          v4i;
typedef __attribute__((ext_vector_type(8)))  int          v8i;

__device__ __forceinline__ unsigned long long umin64(unsigned long long a,
                                                     unsigned long long b) {
  return a < b ? a : b;
}

__global__ __launch_bounds__(256)
void knn_rdf_kernel(const float* __restrict__ pos, float* __restrict__ out)
{
  // 16 rows x 4096 cols of f32 d^2 (256 KB) + staged positions (48 KB)
  // = ~304 KB LDS; CDNA5 WGP has 320 KB.
  __shared__ float s_d2[TILE * NPTS];
  __shared__ float s_pos[NPTS * 3];
  __shared__ float s_sqA[TILE];

  const int bid  = blockIdx.x;
  const int b    = bid >> 8;        // bid / NTILES
  const int it   = bid & (NTILES - 1);
  const int i0   = it * TILE;

  const int tid  = threadIdx.x;
  const int lane = tid & 31;
  const int wave = tid >> 5;
  const int half = lane >> 4;       // 0: lanes 0-15, 1: lanes 16-31
  const int l15  = lane & 15;

  const float* posb = pos + (size_t)b * NPTS * 3;

  // ---- Stage pos[b] (4096x3 f32 = 48 KB) into LDS via the Tensor Data Mover ----
#if __has_builtin(__builtin_amdgcn_tensor_load_to_lds)
  if (wave == 0) {
    const unsigned lds_off = (unsigned)(size_t)(void*)&s_pos[0]; // LDS byte offset
    const unsigned long long ga = (unsigned long long)(size_t)posb;

    // D# group 0: count=1 | lds_addr | global_addr[56:0] | type=2 ("image")
    v4u g0;
    g0[0] = 1u;                                     // count=1, user descriptor
    g0[1] = lds_off;                                // lds_addr (bytes)
    g0[2] = (unsigned)(ga & 0xFFFFFFFFu);           // global_addr[31:0]
    g0[3] = (unsigned)((ga >> 32) & 0x01FFFFFFu)    // global_addr[56:32]
          | (2u << 30);                             // type = 2
    // D# group 1: flat 1-row tensor: 12288 x 1 elements of 4 bytes
    v8i g1;
    g1[0] = (int)(2u << 16);          // wg_mask=0 | data_size=2 (4 bytes)
    g1[1] = (int)(12288u << 16);      // tensor_dim0[15:0] (bits 63:48)
    g1[2] = (int)(1u << 16);          // tensor_dim0 hi = 0 | tensor_dim1 = 1
    g1[3] = (int)(12288u << 16);      // tensor_dim1 hi = 0 | tile_dim0 = 12288
    g1[4] = 1;                        // tile_dim1 = 1, tile_dim2 = 0
    g1[5] = 12288;                    // tensor_dim0_stride[31:0]
    g1[6] = (int)(12288u << 16);      // stride0 hi = 0 | tensor_dim1_stride lo
    g1[7] = 0;                        // tensor_dim1_stride hi
    v4i gz = {};
#if defined(__clang_major__) && (__clang_major__ >= 23)
    v8i gz8 = {};
    __builtin_amdgcn_tensor_load_to_lds(g0, g1, gz, gz, gz8, 0);
#else
    __builtin_amdgcn_tensor_load_to_lds(g0, g1, gz, gz, 0);
#endif
    __builtin_amdgcn_s_wait_tensorcnt(0);           // TENSORcnt -> 0
  }
  __syncthreads();
#else
  for (int i = tid; i < NPTS * 3; i += 256) s_pos[i] = posb[i];
  __syncthreads();
#endif

  // --- squared norms of the 16 A-rows (from LDS) ---
  if (tid < TILE) {
    const float* p = &s_pos[(i0 + tid) * 3];
    s_sqA[tid] = p[0] * p[0] + p[1] * p[1] + p[2] * p[2];
  }
  __syncthreads();

  // ---------------- Phase 1: WMMA pairwise distances into LDS ----------------
  // A matrix (16x4 f32): lanes 0-15 hold M=lane, K=0(x),1(y); lanes 16-31 hold
  // M=lane-16, K=2(z),3(pad=0).  One wmma per 16x16 output tile.
  {
    const float* pa = &s_pos[(i0 + l15) * 3];
    v2f A;
    if (half == 0) { A.x = pa[0]; A.y = pa[1]; }
    else           { A.x = pa[2]; A.y = 0.0f; }

    float sqa[8];
#pragma unroll
    for (int r = 0; r < 8; ++r) sqa[r] = s_sqA[half * 8 + r];

    const float INF = __uint_as_float(0x7F800000u);

    for (int t = 0; t < NTILES / WAVES; ++t) {
      const int jt = wave * (NTILES / WAVES) + t;
      const int j0 = jt * TILE;
      const int gj = j0 + l15;
      const float* pb = &s_pos[gj * 3];
      const float qx = pb[0], qy = pb[1], qz = pb[2];
      const float sqb = qx * qx + qy * qy + qz * qz;

      // B matrix (4x16 f32): lanes 0-15 hold N=lane, K=0,1; lanes 16-31 K=2,3
      v2f Bv;
      if (half == 0) { Bv.x = qx; Bv.y = qy; }
      else           { Bv.x = qz; Bv.y = 0.0f; }
      v8f C = {};
      C = __builtin_amdgcn_wmma_f32_16x16x4_f32(
              false, A, false, Bv, (short)0, C, false, false);

#pragma unroll
      for (int r = 0; r < 8; ++r) {
        const int rowm = half * 8 + r;   // C layout: lanes<16 -> M=r, else M=r+8
        const int gi   = i0 + rowm;
        float d2 = sqa[r] + sqb - 2.0f * C[r];
        d2 = fmaxf(d2, 0.0f);            // keep float bits order-monotone
        if (gi == gj) d2 = INF;          // no self loops
        s_d2[rowm * NPTS + gj] = d2;
      }
    }
  }
  __syncthreads();

  // ---------------- Phase 2: per-row top-50 from LDS + RDF ----------------
  // Each wave owns 2 rows. Per-lane strided min keys: (d2_bits<<32)|col so
  // ordering is ascending d2 with lowest-index tie-break (matches top_k).
  const float INF = __uint_as_float(0x7F800000u);
  for (int rr = 0; rr < 2; ++rr) {
    const int rowm = wave * 2 + rr;
    const int gi   = i0 + rowm;
    float* row = &s_d2[rowm * NPTS];

    unsigned long long key = ~0ull;
    for (int j = lane; j < NPTS; j += 32) {
      unsigned long long k2 =
          ((unsigned long long)__float_as_uint(row[j]) << 32) | (unsigned)j;
      key = umin64(key, k2);
    }

    for (int k = 0; k < KNN; ++k) {
      // wave32 min-reduce (broadcasts result to all lanes)
      unsigned long long m = key;
#pragma unroll
      for (int off = 16; off > 0; off >>= 1)
        m = umin64(m, (unsigned long long)__shfl_xor(m, off, 32));

      const unsigned idx = (unsigned)(m & 0xFFFFFFFFu);
      const float    d2  = __uint_as_float((unsigned)(m >> 32));

      // owning lane retires this element and rescans its stride
      if ((idx & 31u) == (unsigned)lane) {
        row[idx] = INF;
        unsigned long long nk = ~0ull;
        for (int j = lane; j < NPTS; j += 32) {
          unsigned long long k2 =
              ((unsigned long long)__float_as_uint(row[j]) << 32) | (unsigned)j;
          nk = umin64(nk, k2);
        }
        key = nk;
      }

      // one lane writes edge + RDF for this k
      if (lane == (k & 31)) {
        const size_t NK   = (size_t)NPTS * KNN;
        const size_t eoff = (size_t)b * 2 * NK + (size_t)gi * KNN + k;
        out[eoff]      = (float)idx;   // edge_index[b][0][...] = neighbor j
        out[eoff + NK] = (float)gi;    // edge_index[b][1][...] = center i

        const float r = sqrtf(fmaxf(d2, 0.0f));
        const size_t roff =
            (size_t)BATCH * 2 * NK + (((size_t)b * NPTS + gi) * KNN + k) * NBINS;
#pragma unroll
        for (int mb = 0; mb < NBINS; ++mb) {
          const float mu = 2.5f * (float)mb;   // linspace(0,10,5)
          const float dd = r - mu;
          out[roff + mb] = __expf(-0.5f * dd * dd);  // gamma = 0.5
        }
      }
    }
  }
}

extern "C" void kernel_launch(void* const* d_in, const int* in_sizes, int n_in,
                              void* d_out, int out_size, void* d_ws, size_t ws_size,
                              hipStream_t stream) {
  (void)in_sizes; (void)n_in; (void)out_size; (void)d_ws; (void)ws_size;
  const float* pos = (const float*)d_in[0];
  float* out = (float*)d_out;
  dim3 grid(BATCH * NTILES);   // 4096 workgroups, one 16-row stripe each
  dim3 block(256);             // 8 wave32
  knn_rdf_kernel<<<grid, block, 0, stream>>>(pos, out);
}